// RaySampler_84585085927654
// MI455X (gfx1250) — compile-verified
//
#include <hip/hip_runtime.h>
#include <math.h>

typedef float v2f __attribute__((ext_vector_type(2)));
typedef float v8f __attribute__((ext_vector_type(8)));

// ---------------------------------------------------------------------------
// Ray directions via V_WMMA_F32_16X16X4_F32.
// world = cam2world (4x4) x cam_rel_point (4-vec) for 16 pixel columns per
// WMMA. A = camera matrix replicated down 16 rows (row m uses cam row m&3),
// so D rows 8..10 (lanes 16..31) duplicate rows 0..2 -> two WMMAs cover 32
// distinct pixels with every lane holding SoA (x,y,z) and no cross-lane ops.
// ---------------------------------------------------------------------------
__global__ void __launch_bounds__(256)
ray_dirs_wmma_kernel(const float* __restrict__ c2w,   // N x 16 (row-major 4x4)
                     const float* __restrict__ intr,  // N x 9  (row-major 3x3)
                     float* __restrict__ dirs,        // N x M x 3
                     unsigned int M, unsigned int res)
{
    const unsigned int cam  = blockIdx.y;
    const unsigned int tid  = threadIdx.x;
    const unsigned int lane = tid & 31u;
    const unsigned int m15  = lane & 15u;
    const bool hi = lane >= 16u;

    const float* Mc = c2w  + (size_t)cam * 16u;
    const float* Ic = intr + (size_t)cam * 9u;

    const float fx = Ic[0], sk = Ic[1], cx = Ic[2];
    const float fy = Ic[4], cy = Ic[5];
    const float lx = Mc[3], ly = Mc[7], lz = Mc[11];   // cam_locs_world

    const unsigned int waveBase = blockIdx.x * blockDim.x + (tid & ~31u);
    const unsigned int p        = waveBase + lane;     // this lane's output pixel

    const float fres  = (float)res;
    const float skofy = sk / fy;
    const float cterm = cy * skofy;                    // cy*sk/fy

    // lifted camera-space coords (z_cam = -1 folded in; matches reference)
    auto lift = [&](unsigned int q, float& xl, float& yl) {
        unsigned int row = q / res;
        unsigned int col = q - row * res;
        float x_cam = ((float)col + 0.5f) / fres;
        float y_cam = ((float)row + 0.5f) / fres;
        xl = (x_cam - cx + cterm - skofy * y_cam) / fx;
        yl = -(y_cam - cy) / fy;
    };

    if (waveBase + 32u <= M) {
        // ---- full wave: EXEC all ones, WMMA path ----
        const unsigned int q0 = waveBase + m15;   // B0 column pixel
        const unsigned int q1 = q0 + 16u;         // B1 column pixel
        float xl0, yl0, xl1, yl1;
        lift(q0, xl0, yl0);
        lift(q1, xl1, yl1);

        // A 16x4 f32 layout: VGPR0 = K0 | K2 (lane halves), VGPR1 = K1 | K3
        const unsigned int r4 = m15 & 3u;
        v2f A;
        A.x = Mc[r4 * 4u + (hi ? 2u : 0u)];
        A.y = Mc[r4 * 4u + (hi ? 3u : 1u)];

        // B 4x16 f32 layout: lanes 0-15 carry K=0,1; lanes 16-31 carry K=2,3
        v2f B0, B1;
        B0.x = hi ? -1.0f : xl0;
        B0.y = hi ?  1.0f : yl0;
        B1.x = hi ? -1.0f : xl1;
        B1.y = hi ?  1.0f : yl1;

        v8f Z = {};
        v8f dA = __builtin_amdgcn_wmma_f32_16x16x4_f32(
            false, A, false, B0, (short)0, Z, false, false);
        v8f dB = __builtin_amdgcn_wmma_f32_16x16x4_f32(
            false, A, false, B1, (short)0, Z, false, false);

        // lanes 0-15: pixel waveBase+lane from dA rows 0..2
        // lanes 16-31: pixel waveBase+lane from dB rows 8..10 (replicated A)
        float wx = hi ? dB[0] : dA[0];
        float wy = hi ? dB[1] : dA[1];
        float wz = hi ? dB[2] : dA[2];

        float dx = wx - lx, dy = wy - ly, dz = wz - lz;
        float inv = 1.0f / fmaxf(sqrtf(dx * dx + dy * dy + dz * dz), 1e-12f);
        float* o = dirs + ((size_t)cam * M + p) * 3u;
        o[0] = dx * inv;
        o[1] = dy * inv;
        o[2] = dz * inv;
    } else if (p < M) {
        // ---- tail (partial wave): plain VALU, EXEC-safe ----
        float xl, yl;
        lift(p, xl, yl);
        // dir = M[:, :3] . (xl, yl, -1)   (the w column cancels with cam_loc)
        float dx = Mc[0] * xl + Mc[1] * yl - Mc[2];
        float dy = Mc[4] * xl + Mc[5] * yl - Mc[6];
        float dz = Mc[8] * xl + Mc[9] * yl - Mc[10];
        float inv = 1.0f / fmaxf(sqrtf(dx * dx + dy * dy + dz * dz), 1e-12f);
        float* o = dirs + ((size_t)cam * M + p) * 3u;
        o[0] = dx * inv;
        o[1] = dy * inv;
        o[2] = dz * inv;
    }
}

// ---------------------------------------------------------------------------
// Ray origins: per-camera constant broadcast, written as rotated float4s
// ((x,y,z,x),(y,z,x,y),(z,x,y,z)) for full 16-byte stores.
// ---------------------------------------------------------------------------
__global__ void __launch_bounds__(256)
ray_origins_vec4_kernel(const float* __restrict__ c2w,
                        float4* __restrict__ out4,
                        unsigned int q_per_cam)
{
    const unsigned int cam = blockIdx.y;
    const unsigned int q   = blockIdx.x * blockDim.x + threadIdx.x;
    if (q >= q_per_cam) return;
    const float* Mc = c2w + (size_t)cam * 16u;
    const float x = Mc[3], y = Mc[7], z = Mc[11];
    const unsigned int r = q % 3u;
    float4 v = (r == 0u) ? make_float4(x, y, z, x)
             : (r == 1u) ? make_float4(y, z, x, y)
                         : make_float4(z, x, y, z);
    out4[(size_t)cam * q_per_cam + q] = v;
}

__global__ void __launch_bounds__(256)
ray_origins_scalar_kernel(const float* __restrict__ c2w,
                          float* __restrict__ out,
                          unsigned int M)
{
    const unsigned int cam = blockIdx.y;
    const unsigned int p   = blockIdx.x * blockDim.x + threadIdx.x;
    if (p >= M) return;
    const float* Mc = c2w + (size_t)cam * 16u;
    float* o = out + ((size_t)cam * M + p) * 3u;
    o[0] = Mc[3];
    o[1] = Mc[7];
    o[2] = Mc[11];
}

extern "C" void kernel_launch(void* const* d_in, const int* in_sizes, int n_in,
                              void* d_out, int out_size, void* d_ws, size_t ws_size,
                              hipStream_t stream) {
    const float* c2w  = (const float*)d_in[0];   // (N,4,4) f32
    const float* intr = (const float*)d_in[1];   // (N,3,3) f32
    // d_in[2] = resolution (device scalar) -- derive dims on host instead:
    const int N = in_sizes[0] / 16;
    const unsigned long long M64 = (unsigned long long)out_size / (6ULL * (unsigned)N);
    const unsigned int M   = (unsigned int)M64;
    const unsigned int res = (unsigned int)(sqrt((double)M64) + 0.5);

    float* out     = (float*)d_out;
    float* origins = out;                                   // (N,M,3)
    float* dirs    = out + (size_t)N * M * 3u;              // (N,M,3)

    // ---- origins (pure broadcast stores) ----
    const unsigned long long ofloats = (unsigned long long)M * 3ULL;
    if ((ofloats & 3ULL) == 0ULL) {
        const unsigned int q_per_cam = (unsigned int)(ofloats >> 2);  // float4s per camera
        dim3 grid((q_per_cam + 255u) / 256u, N);
        ray_origins_vec4_kernel<<<grid, 256, 0, stream>>>(
            c2w, (float4*)origins, q_per_cam);
    } else {
        dim3 grid((M + 255u) / 256u, N);
        ray_origins_scalar_kernel<<<grid, 256, 0, stream>>>(c2w, origins, M);
    }

    // ---- directions (WMMA einsum + normalize) ----
    {
        dim3 grid((M + 255u) / 256u, N);
        ray_dirs_wmma_kernel<<<grid, 256, 0, stream>>>(c2w, intr, dirs, M, res);
    }
}